// Decoder_16716012716062
// MI455X (gfx1250) — compile-verified
//
#include <hip/hip_runtime.h>
#include <math.h>

// Problem constants (from reference)
#define VCAB  50000
#define EDIM  128
#define HDIM  256
#define SLEN  400
#define BATCH 128
#define OOVN  50
#define H2    512          // 2*H
#define VO    (VCAB+OOVN)  // 50050

typedef float v2f_t __attribute__((ext_vector_type(2)));
typedef float v8f_t __attribute__((ext_vector_type(8)));

__device__ __forceinline__ float sigm(float x) { return 1.0f / (1.0f + expf(-x)); }

// ---------------------------------------------------------------------------
// K1: embedding + input proj + LSTM cell + dec_feat = status @ att_ws^T
// grid = (B), block = 256
// ---------------------------------------------------------------------------
__global__ __launch_bounds__(256) void lstm_kernel(
    const int* __restrict__ tok, const float* __restrict__ h0,
    const float* __restrict__ c0, const float* __restrict__ ctxvec,
    const float* __restrict__ emb, const float* __restrict__ Wi,
    const float* __restrict__ bi, const float* __restrict__ W_ih,
    const float* __restrict__ W_hh, const float* __restrict__ b_ih,
    const float* __restrict__ b_hh, const float* __restrict__ att_ws,
    float* __restrict__ out_h, float* __restrict__ out_c,
    float* __restrict__ ws_x, float* __restrict__ ws_dec)
{
    __shared__ float sCat[H2 + EDIM];   // [context(512), embed(128)]
    __shared__ float sX[EDIM];
    __shared__ float sH0[HDIM];
    __shared__ float sGates[4 * HDIM];
    __shared__ float sHC[H2];           // [h_t, c_t]

    const int b = blockIdx.x, t = threadIdx.x;
    const int tk = tok[b];
    for (int k = t; k < H2; k += 256)   sCat[k] = ctxvec[b * H2 + k];
    if (t < EDIM)                       sCat[H2 + t] = emb[(size_t)tk * EDIM + t];
    if (t < HDIM)                       sH0[t] = h0[b * HDIM + t];
    __syncthreads();

    if (t < EDIM) {
        float acc = bi[t];
        const float* w = Wi + (size_t)t * (H2 + EDIM);
        for (int k = 0; k < H2 + EDIM; ++k) acc += sCat[k] * w[k];
        sX[t] = acc;
        ws_x[b * EDIM + t] = acc;
    }
    __syncthreads();

    for (int j = t; j < 4 * HDIM; j += 256) {
        float acc = b_ih[j] + b_hh[j];
        const float* wa = W_ih + (size_t)j * EDIM;
        for (int k = 0; k < EDIM; ++k) acc += sX[k] * wa[k];
        const float* wb = W_hh + (size_t)j * HDIM;
        for (int k = 0; k < HDIM; ++k) acc += sH0[k] * wb[k];
        sGates[j] = acc;
    }
    __syncthreads();

    {   // torch gate order i,f,g,o
        const int j = t;  // t in [0,256)
        float gi = sGates[j], gf = sGates[HDIM + j];
        float gg = sGates[2 * HDIM + j], go = sGates[3 * HDIM + j];
        float cc = sigm(gf) * c0[b * HDIM + j] + sigm(gi) * tanhf(gg);
        float hh = sigm(go) * tanhf(cc);
        out_h[b * HDIM + j] = hh;
        out_c[b * HDIM + j] = cc;
        sHC[j] = hh;
        sHC[HDIM + j] = cc;
    }
    __syncthreads();

    for (int o = t; o < H2; o += 256) {
        float acc = 0.f;
        const float* w = att_ws + (size_t)o * H2;
        for (int k = 0; k < H2; ++k) acc += sHC[k] * w[k];
        ws_dec[b * H2 + o] = acc;
    }
}

// ---------------------------------------------------------------------------
// K2: e_t = tanh(enc @ att_wh^T + dec_feat + cov*wc + bc) . att_v  (masked)
// One block = 16 (b,s) rows (contiguous; 400/16=25 so never crosses batch).
// 8 waves; each wave WMMAs 4 column tiles of 16 over K=512.
// grid = (B*S/16 = 3200), block = 256
// ---------------------------------------------------------------------------
__global__ __launch_bounds__(256) void et_wmma_kernel(
    const float* __restrict__ enc, const float* __restrict__ att_wh,
    const float* __restrict__ att_wc, const float* __restrict__ att_bc,
    const float* __restrict__ att_v, const float* __restrict__ coverage,
    const int* __restrict__ mask, const float* __restrict__ ws_dec,
    float* __restrict__ ws_et)
{
    __shared__ __align__(16) float As[16 * H2];  // 32 KB encoder tile
    __shared__ float sDec[H2];
    __shared__ float sCov[16];
    __shared__ float sEsum[16];

    const int t = threadIdx.x;
    const int b  = blockIdx.x / 25;
    const int s0 = (blockIdx.x % 25) * 16;

    // cooperative coalesced load of 16 contiguous rows (8192 floats)
    const float4* src = (const float4*)(enc + ((size_t)(b * SLEN + s0)) * H2);
    float4* dst = (float4*)As;
    for (int i = t; i < 16 * H2 / 4; i += 256) dst[i] = src[i];
    for (int k = t; k < H2; k += 256) sDec[k] = ws_dec[b * H2 + k];
    if (t < 16) { sCov[t] = coverage[b * SLEN + s0 + t]; sEsum[t] = 0.f; }
    __syncthreads();

    const int lane  = t & 31;
    const int wave  = t >> 5;
    const int mrow  = lane & 15;
    const int khalf = (lane >> 4) << 1;      // 0 or 2
    const int ncl   = lane & 15;
    const int mhalf = (lane >> 4) << 3;      // 0 or 8

    for (int ti = 0; ti < 4; ++ti) {
        const int ct = wave * 4 + ti;        // column tile in [0,32)
        const int n  = ct * 16 + ncl;        // output feature column
        v8f_t acc = {};
        const float* bcol = att_wh + (size_t)n * H2;   // B[k][n] = att_wh[n][k]
        for (int k0 = 0; k0 < H2; k0 += 4) {
            const int kb = k0 + khalf;
            v2f_t a; a.x = As[mrow * H2 + kb]; a.y = As[mrow * H2 + kb + 1];
            v2f_t bb; bb.x = bcol[kb]; bb.y = bcol[kb + 1];
            acc = __builtin_amdgcn_wmma_f32_16x16x4_f32(
                false, a, false, bb, (short)0, acc, false, false);
        }
        // fused epilogue: tanh(.) * att_v[n], reduce over n into per-row sum
        const float wc = att_wc[n], bc = att_bc[n], av = att_v[n], dc = sDec[n];
        #pragma unroll
        for (int v = 0; v < 8; ++v) {
            const int m = v + mhalf;
            float val = acc[v] + dc + sCov[m] * wc + bc;
            atomicAdd(&sEsum[m], tanhf(val) * av);
        }
    }
    __syncthreads();

    if (t < 16) {
        const int s = s0 + t;
        float ev = (mask[b * SLEN + s] == 0) ? -INFINITY : sEsum[t];
        ws_et[b * SLEN + s] = ev;
    }
}

// ---------------------------------------------------------------------------
// K3: softmax over S, a_t, next_coverage, ctx, gen_p.  grid=(B), block=256
// ---------------------------------------------------------------------------
__global__ __launch_bounds__(256) void softmax_ctx_kernel(
    const float* __restrict__ ws_et, const float* __restrict__ coverage,
    const float* __restrict__ enc, const float* __restrict__ out_h,
    const float* __restrict__ out_c, const float* __restrict__ ws_x,
    const float* __restrict__ gp_wh, const float* __restrict__ gp_bh,
    const float* __restrict__ gp_ws, const float* __restrict__ gp_bs,
    const float* __restrict__ gp_wx, const float* __restrict__ gp_bx,
    float* __restrict__ out_at, float* __restrict__ out_cov,
    float* __restrict__ out_ctx, float* __restrict__ out_gp)
{
    __shared__ float sE[SLEN];
    __shared__ float sA[SLEN];
    __shared__ float sCtx[H2];
    __shared__ float sRed[256];

    const int b = blockIdx.x, t = threadIdx.x;
    for (int s = t; s < SLEN; s += 256) sE[s] = ws_et[b * SLEN + s];
    __syncthreads();

    float mx = -INFINITY;
    for (int s = t; s < SLEN; s += 256) mx = fmaxf(mx, sE[s]);
    sRed[t] = mx; __syncthreads();
    for (int off = 128; off > 0; off >>= 1) {
        if (t < off) sRed[t] = fmaxf(sRed[t], sRed[t + off]);
        __syncthreads();
    }
    mx = sRed[0]; __syncthreads();

    float sm = 0.f;
    for (int s = t; s < SLEN; s += 256) sm += expf(sE[s] - mx);
    sRed[t] = sm; __syncthreads();
    for (int off = 128; off > 0; off >>= 1) {
        if (t < off) sRed[t] += sRed[t + off];
        __syncthreads();
    }
    const float denom = sRed[0]; __syncthreads();

    for (int s = t; s < SLEN; s += 256) {
        float a = expf(sE[s] - mx) / denom;
        sA[s] = a;
        out_at[b * SLEN + s]  = a;
        out_cov[b * SLEN + s] = coverage[b * SLEN + s] + a;
    }
    __syncthreads();

    for (int d = t; d < H2; d += 256) {
        float acc = 0.f;
        const float* ep = enc + (size_t)b * SLEN * H2 + d;
        for (int s = 0; s < SLEN; ++s) acc += sA[s] * ep[(size_t)s * H2];
        sCtx[d] = acc;
        out_ctx[b * H2 + d] = acc;
    }
    __syncthreads();

    float p = 0.f;
    for (int k = t; k < H2; k += 256) {
        p += gp_wh[k] * sCtx[k];
        float st = (k < HDIM) ? out_h[b * HDIM + k] : out_c[b * HDIM + (k - HDIM)];
        p += gp_ws[k] * st;
    }
    for (int k = t; k < EDIM; k += 256) p += gp_wx[k] * ws_x[b * EDIM + k];
    sRed[t] = p; __syncthreads();
    for (int off = 128; off > 0; off >>= 1) {
        if (t < off) sRed[t] += sRed[t + off];
        __syncthreads();
    }
    if (t == 0) out_gp[b] = sigm(sRed[0] + gp_bh[0] + gp_bs[0] + gp_bx[0]);
}

// ---------------------------------------------------------------------------
// K4: hidden = relu([h_t, ctx] @ out_w1^T + b1).  grid=(B), block=256
// ---------------------------------------------------------------------------
__global__ __launch_bounds__(256) void hidden_kernel(
    const float* __restrict__ out_h, const float* __restrict__ out_ctx,
    const float* __restrict__ w1, const float* __restrict__ b1,
    float* __restrict__ ws_hid)
{
    __shared__ float sF[HDIM + H2];  // 768
    const int b = blockIdx.x, t = threadIdx.x;
    if (t < HDIM) sF[t] = out_h[b * HDIM + t];
    for (int k = t; k < H2; k += 256) sF[HDIM + k] = out_ctx[b * H2 + k];
    __syncthreads();

    float acc = b1[t];
    const float* w = w1 + (size_t)t * (HDIM + H2);
    for (int k = 0; k < HDIM + H2; ++k) acc += sF[k] * w[k];
    ws_hid[b * HDIM + t] = fmaxf(acc, 0.f);
}

// ---------------------------------------------------------------------------
// K5: logits = hidden @ out_w2^T + b2 via WMMA f32 16x16x4.
// grid = (3125, 8), block = 32 (one wave per 16x16 tile), K = 256.
// ---------------------------------------------------------------------------
__global__ __launch_bounds__(32) void logits_wmma_kernel(
    const float* __restrict__ hid, const float* __restrict__ w2,
    const float* __restrict__ b2, float* __restrict__ logits)
{
    const int nt = blockIdx.x, mt = blockIdx.y;
    const int lane  = threadIdx.x & 31;
    const int mrow  = lane & 15;
    const int khalf = (lane >> 4) << 1;
    const int n     = nt * 16 + (lane & 15);
    const int mhalf = (lane >> 4) << 3;

    const float* arow = hid + (size_t)(mt * 16 + mrow) * HDIM;
    const float* bcol = w2 + (size_t)n * HDIM;   // B[k][n] = out_w2[n][k]
    v8f_t acc = {};
    for (int k0 = 0; k0 < HDIM; k0 += 4) {
        const int kb = k0 + khalf;
        v2f_t a; a.x = arow[kb]; a.y = arow[kb + 1];
        v2f_t bb; bb.x = bcol[kb]; bb.y = bcol[kb + 1];
        acc = __builtin_amdgcn_wmma_f32_16x16x4_f32(
            false, a, false, bb, (short)0, acc, false, false);
    }
    const float bias = b2[n];
    #pragma unroll
    for (int v = 0; v < 8; ++v) {
        const int m = v + mhalf;
        logits[(size_t)(mt * 16 + m) * VCAB + n] = acc[v] + bias;
    }
}

// ---------------------------------------------------------------------------
// K6: per-row max and sum(exp) over V.  grid=(B), block=256
// ---------------------------------------------------------------------------
__global__ __launch_bounds__(256) void rowstats_kernel(
    const float* __restrict__ logits, float* __restrict__ stats)
{
    __shared__ float sRed[256];
    const int b = blockIdx.x, t = threadIdx.x;
    const float* row = logits + (size_t)b * VCAB;

    float mx = -INFINITY;
    for (int j = t; j < VCAB; j += 256) mx = fmaxf(mx, row[j]);
    sRed[t] = mx; __syncthreads();
    for (int off = 128; off > 0; off >>= 1) {
        if (t < off) sRed[t] = fmaxf(sRed[t], sRed[t + off]);
        __syncthreads();
    }
    mx = sRed[0]; __syncthreads();

    float sm = 0.f;
    for (int j = t; j < VCAB; j += 256) sm += expf(row[j] - mx);
    sRed[t] = sm; __syncthreads();
    for (int off = 128; off > 0; off >>= 1) {
        if (t < off) sRed[t] += sRed[t + off];
        __syncthreads();
    }
    if (t == 0) { stats[2 * b] = mx; stats[2 * b + 1] = sRed[0]; }
}

// ---------------------------------------------------------------------------
// K7: final_dist = [softmax(logits)*gen_p, oovs_zero]
// ---------------------------------------------------------------------------
__global__ __launch_bounds__(256) void final_fill_kernel(
    const float* __restrict__ logits, const float* __restrict__ stats,
    const float* __restrict__ out_gp, const float* __restrict__ oovs,
    float* __restrict__ fin)
{
    const int idx = blockIdx.x * 256 + threadIdx.x;
    if (idx >= BATCH * VO) return;
    const int b = idx / VO;
    const int j = idx - b * VO;
    float v;
    if (j < VCAB) {
        v = expf(logits[(size_t)b * VCAB + j] - stats[2 * b]) / stats[2 * b + 1]
            * out_gp[b];
    } else {
        v = oovs[b * OOVN + (j - VCAB)];
    }
    fin[idx] = v;
}

// ---------------------------------------------------------------------------
// K8: pointer scatter-add of a_t*(1-gen_p) at encoder_with_oov
// ---------------------------------------------------------------------------
__global__ __launch_bounds__(256) void scatter_kernel(
    const int* __restrict__ ewo, const float* __restrict__ out_at,
    const float* __restrict__ out_gp, float* __restrict__ fin)
{
    const int idx = blockIdx.x * 256 + threadIdx.x;
    if (idx >= BATCH * SLEN) return;
    const int b = idx / SLEN;
    const int col = ewo[idx];
    atomicAdd(&fin[(size_t)b * VO + col], out_at[idx] * (1.f - out_gp[b]));
}

// ---------------------------------------------------------------------------
extern "C" void kernel_launch(void* const* d_in, const int* in_sizes, int n_in,
                              void* d_out, int out_size, void* d_ws, size_t ws_size,
                              hipStream_t stream) {
    const int*   tok    = (const int*)  d_in[0];
    const float* h0     = (const float*)d_in[1];
    const float* c0     = (const float*)d_in[2];
    const float* enc    = (const float*)d_in[3];
    const int*   mask   = (const int*)  d_in[4];
    const float* ctxv   = (const float*)d_in[5];
    const float* oovs   = (const float*)d_in[6];
    const int*   ewo    = (const int*)  d_in[7];
    const float* cov    = (const float*)d_in[8];
    const float* emb    = (const float*)d_in[9];
    const float* Wi     = (const float*)d_in[10];
    const float* bi     = (const float*)d_in[11];
    const float* W_ih   = (const float*)d_in[12];
    const float* W_hh   = (const float*)d_in[13];
    const float* b_ih   = (const float*)d_in[14];
    const float* b_hh   = (const float*)d_in[15];
    const float* att_wh = (const float*)d_in[16];
    const float* att_ws = (const float*)d_in[17];
    const float* att_wc = (const float*)d_in[18];
    const float* att_bc = (const float*)d_in[19];
    const float* att_v  = (const float*)d_in[20];
    const float* gp_wh  = (const float*)d_in[21];
    const float* gp_bh  = (const float*)d_in[22];
    const float* gp_ws  = (const float*)d_in[23];
    const float* gp_bs  = (const float*)d_in[24];
    const float* gp_wx  = (const float*)d_in[25];
    const float* gp_bx  = (const float*)d_in[26];
    const float* out_w1 = (const float*)d_in[27];
    const float* out_b1 = (const float*)d_in[28];
    const float* out_w2 = (const float*)d_in[29];
    const float* out_b2 = (const float*)d_in[30];

    float* out = (float*)d_out;
    float* o_final = out;                                  // B*VO
    float* o_h     = o_final + (size_t)BATCH * VO;         // B*H
    float* o_c     = o_h + BATCH * HDIM;                   // B*H
    float* o_ctx   = o_c + BATCH * HDIM;                   // B*2H
    float* o_at    = o_ctx + BATCH * H2;                   // B*S
    float* o_gp    = o_at + BATCH * SLEN;                  // B
    float* o_cov   = o_gp + BATCH;                         // B*S

    float* wsf       = (float*)d_ws;
    float* ws_x      = wsf;                                // B*E     = 16384
    float* ws_dec    = ws_x + BATCH * EDIM;                // B*2H    = 65536
    float* ws_et     = ws_dec + BATCH * H2;                // B*S     = 51200
    float* ws_hid    = ws_et + BATCH * SLEN;               // B*H     = 32768
    float* ws_stats  = ws_hid + BATCH * HDIM;              // 2*B     = 256
    float* ws_logits = ws_stats + 2 * BATCH;               // B*V     = 6.4M

    lstm_kernel<<<BATCH, 256, 0, stream>>>(
        tok, h0, c0, ctxv, emb, Wi, bi, W_ih, W_hh, b_ih, b_hh, att_ws,
        o_h, o_c, ws_x, ws_dec);

    et_wmma_kernel<<<BATCH * SLEN / 16, 256, 0, stream>>>(
        enc, att_wh, att_wc, att_bc, att_v, cov, mask, ws_dec, ws_et);

    softmax_ctx_kernel<<<BATCH, 256, 0, stream>>>(
        ws_et, cov, enc, o_h, o_c, ws_x,
        gp_wh, gp_bh, gp_ws, gp_bs, gp_wx, gp_bx,
        o_at, o_cov, o_ctx, o_gp);

    hidden_kernel<<<BATCH, 256, 0, stream>>>(o_h, o_ctx, out_w1, out_b1, ws_hid);

    logits_wmma_kernel<<<dim3(VCAB / 16, BATCH / 16), 32, 0, stream>>>(
        ws_hid, out_w2, out_b2, ws_logits);

    rowstats_kernel<<<BATCH, 256, 0, stream>>>(ws_logits, ws_stats);

    final_fill_kernel<<<(BATCH * VO + 255) / 256, 256, 0, stream>>>(
        ws_logits, ws_stats, o_gp, oovs, o_final);

    scatter_kernel<<<(BATCH * SLEN + 255) / 256, 256, 0, stream>>>(
        ewo, o_at, o_gp, o_final);
}